// ControlUnit_13125420057183
// MI455X (gfx1250) — compile-verified
//
#include <hip/hip_runtime.h>
#include <math.h>

typedef float v2f __attribute__((ext_vector_type(2)));
typedef float v4f __attribute__((ext_vector_type(4)));
typedef float v8f __attribute__((ext_vector_type(8)));

#define DIM_     512
#define TWO_DIM  1024
#define B_       256
#define S_       512

// ---------------------------------------------------------------------------
// GEMM1: p[B,D] = q[B,2D] @ pos_w[t][D,2D]^T + pos_b[t]
// One 16x16 output tile per wave, K-loop in steps of 4 using
// V_WMMA_F32_16X16X4_F32.  A-frag: lane(half,lr) holds A[lr, k+2*half .. +1]
// as a float2; B-frag symmetric (B[kk,n] = pos_w[n0+n, k+kk]).
// ---------------------------------------------------------------------------
__global__ __launch_bounds__(128) void gemm_pos(const int* __restrict__ tptr,
                                                const float* __restrict__ q,
                                                const float* __restrict__ pos_w,
                                                const float* __restrict__ pos_b,
                                                float* __restrict__ p_out) {
    const int t = tptr[0];
    const float* pw = pos_w + (size_t)t * DIM_ * TWO_DIM;
    const float* pb = pos_b + (size_t)t * DIM_;

    const int lane = threadIdx.x & 31;
    const int wave = threadIdx.x >> 5;
    const int tile = blockIdx.x * 4 + wave;     // 512 tiles total
    const int m0 = (tile >> 5) * 16;            // 16 M-tiles
    const int n0 = (tile & 31) * 16;            // 32 N-tiles

    const int half = lane >> 4;                 // 0: K=0,1   1: K=2,3
    const int lr   = lane & 15;                 // A row / B col within tile
    const int kb   = half * 2;

    const float* Abase = q  + (size_t)(m0 + lr) * TWO_DIM + kb;
    const float* Bbase = pw + (size_t)(n0 + lr) * TWO_DIM + kb;

    v8f acc = {0.f, 0.f, 0.f, 0.f, 0.f, 0.f, 0.f, 0.f};
#pragma unroll 8
    for (int k = 0; k < TWO_DIM; k += 4) {
        v2f a = *(const v2f*)(Abase + k);
        v2f b = *(const v2f*)(Bbase + k);
        acc = __builtin_amdgcn_wmma_f32_16x16x4_f32(false, a, false, b,
                                                    (short)0, acc, false, false);
    }

    const float bias = pb[n0 + lr];
#pragma unroll
    for (int v = 0; v < 8; ++v) {
        // C/D layout: lane(half,lr), vgpr v -> row m0+v+8*half, col n0+lr
        p_out[(size_t)(m0 + v + 8 * half) * DIM_ + n0 + lr] = acc[v] + bias;
    }
}

// ---------------------------------------------------------------------------
// GEMM2: w_eff[B,D] = ((concat(c,p) @ cq_w^T) + cq_b) * attn_w
// Same tiling; K-loop split over the two concat halves (uniform branch-free).
// ---------------------------------------------------------------------------
__global__ __launch_bounds__(128) void gemm_cq(const float* __restrict__ c,
                                               const float* __restrict__ p,
                                               const float* __restrict__ cq_w,
                                               const float* __restrict__ cq_b,
                                               const float* __restrict__ attn_w,
                                               float* __restrict__ weff) {
    const int lane = threadIdx.x & 31;
    const int wave = threadIdx.x >> 5;
    const int tile = blockIdx.x * 4 + wave;
    const int m0 = (tile >> 5) * 16;
    const int n0 = (tile & 31) * 16;

    const int half = lane >> 4;
    const int lr   = lane & 15;
    const int kb   = half * 2;

    const float* Ac    = c    + (size_t)(m0 + lr) * DIM_ + kb;
    const float* Ap    = p    + (size_t)(m0 + lr) * DIM_ + kb;
    const float* Bbase = cq_w + (size_t)(n0 + lr) * TWO_DIM + kb;

    v8f acc = {0.f, 0.f, 0.f, 0.f, 0.f, 0.f, 0.f, 0.f};
#pragma unroll 8
    for (int k = 0; k < DIM_; k += 4) {          // K = 0..511 : A = c
        v2f a = *(const v2f*)(Ac + k);
        v2f b = *(const v2f*)(Bbase + k);
        acc = __builtin_amdgcn_wmma_f32_16x16x4_f32(false, a, false, b,
                                                    (short)0, acc, false, false);
    }
#pragma unroll 8
    for (int k = 0; k < DIM_; k += 4) {          // K = 512..1023 : A = p
        v2f a = *(const v2f*)(Ap + k);
        v2f b = *(const v2f*)(Bbase + DIM_ + k);
        acc = __builtin_amdgcn_wmma_f32_16x16x4_f32(false, a, false, b,
                                                    (short)0, acc, false, false);
    }

    const float bias = cq_b[n0 + lr];
    const float aw   = attn_w[n0 + lr];
#pragma unroll
    for (int v = 0; v < 8; ++v) {
        weff[(size_t)(m0 + v + 8 * half) * DIM_ + n0 + lr] = (acc[v] + bias) * aw;
    }
}

// ---------------------------------------------------------------------------
// Kernel 3: single-pass online softmax + weighted context sum.
// One block per batch row; 16 waves each own S/16 = 32 rows; lane owns
// D/32 = 16 channels (4x b128).  ctx is streamed from HBM exactly once with
// non-temporal cache policy (no reuse -> don't pollute WGP$/L2).
// ---------------------------------------------------------------------------
__global__ __launch_bounds__(512) void attn_ctrl(const float* __restrict__ ctx,
                                                 const float* __restrict__ weff,
                                                 const float* __restrict__ attn_b,
                                                 float* __restrict__ out) {
    const int b    = blockIdx.x;
    const int lane = threadIdx.x & 31;
    const int wave = threadIdx.x >> 5;           // 0..15
    const float ab = attn_b[0];

    // Per-lane effective weights for channels d = lane*16 .. lane*16+15
    const v4f* wrow = (const v4f*)(weff + (size_t)b * DIM_) + lane * 4;
    v4f w4[4];
#pragma unroll
    for (int i = 0; i < 4; ++i) w4[i] = wrow[i];

    float m = -INFINITY, denom = 0.f;
    v4f acc[4];
#pragma unroll
    for (int i = 0; i < 4; ++i) acc[i] = (v4f){0.f, 0.f, 0.f, 0.f};

    const float* ctxb = ctx + (size_t)b * S_ * DIM_;
    for (int s = wave; s < S_; s += 16) {
        const v4f* row = (const v4f*)(ctxb + (size_t)s * DIM_) + lane * 4;
        v4f x[4];
#pragma unroll
        for (int i = 0; i < 4; ++i) x[i] = __builtin_nontemporal_load(row + i);

        float partial = 0.f;
#pragma unroll
        for (int i = 0; i < 4; ++i) {
            partial = fmaf(w4[i].x, x[i].x, partial);
            partial = fmaf(w4[i].y, x[i].y, partial);
            partial = fmaf(w4[i].z, x[i].z, partial);
            partial = fmaf(w4[i].w, x[i].w, partial);
        }
#pragma unroll
        for (int off = 16; off > 0; off >>= 1)
            partial += __shfl_xor(partial, off, 32);

        const float logit = partial + ab;
        const float mn = fmaxf(m, logit);
        const float sc = __expf(m - mn);         // 0 on first iteration
        const float e  = __expf(logit - mn);
        denom = fmaf(denom, sc, e);
#pragma unroll
        for (int i = 0; i < 4; ++i) {
            acc[i].x = fmaf(acc[i].x, sc, e * x[i].x);
            acc[i].y = fmaf(acc[i].y, sc, e * x[i].y);
            acc[i].z = fmaf(acc[i].z, sc, e * x[i].z);
            acc[i].w = fmaf(acc[i].w, sc, e * x[i].w);
        }
        m = mn;
    }

    // Merge the 16 per-wave online-softmax states through LDS (32 KB).
    __shared__ float s_m[16];
    __shared__ float s_d[16];
    __shared__ float s_acc[16 * DIM_];
    float* sa = s_acc + wave * DIM_ + lane * 16;
#pragma unroll
    for (int i = 0; i < 4; ++i) {
        sa[4 * i + 0] = acc[i].x;
        sa[4 * i + 1] = acc[i].y;
        sa[4 * i + 2] = acc[i].z;
        sa[4 * i + 3] = acc[i].w;
    }
    if (lane == 0) { s_m[wave] = m; s_d[wave] = denom; }
    __syncthreads();

    const int d = threadIdx.x;                   // 512 threads == D
    float gm = -INFINITY;
#pragma unroll
    for (int w = 0; w < 16; ++w) gm = fmaxf(gm, s_m[w]);
    float tot = 0.f, val = 0.f;
#pragma unroll
    for (int w = 0; w < 16; ++w) {
        const float f = __expf(s_m[w] - gm);
        tot = fmaf(s_d[w], f, tot);
        val = fmaf(s_acc[w * DIM_ + d], f, val);
    }
    out[(size_t)b * DIM_ + d] = val / tot;
}

// ---------------------------------------------------------------------------
extern "C" void kernel_launch(void* const* d_in, const int* in_sizes, int n_in,
                              void* d_out, int out_size, void* d_ws, size_t ws_size,
                              hipStream_t stream) {
    const int*   t      = (const int*)d_in[0];
    const float* ctx    = (const float*)d_in[1];
    const float* q      = (const float*)d_in[2];
    const float* c      = (const float*)d_in[3];
    const float* pos_w  = (const float*)d_in[4];
    const float* pos_b  = (const float*)d_in[5];
    const float* cq_w   = (const float*)d_in[6];
    const float* cq_b   = (const float*)d_in[7];
    const float* attn_w = (const float*)d_in[8];
    const float* attn_b = (const float*)d_in[9];
    float* out  = (float*)d_out;

    float* p_ws = (float*)d_ws;                      // [B, D]   512 KB
    float* weff = p_ws + (size_t)B_ * DIM_;          // [B, D]   512 KB

    // (B/16)*(D/16) = 512 wave-tiles, 4 waves per 128-thread block
    gemm_pos<<<128, 128, 0, stream>>>(t, q, pos_w, pos_b, p_ws);
    gemm_cq <<<128, 128, 0, stream>>>(c, p_ws, cq_w, cq_b, attn_w, weff);
    attn_ctrl<<<B_, 512, 0, stream>>>(ctx, weff, attn_b, out);
}